// RegionProposalNetwork_644245095178
// MI455X (gfx1250) — compile-verified
//
#include <hip/hip_runtime.h>
#include <math.h>
#include <stdint.h>

#define NIMG 4
#define NLEV 5
#define KTOT 4507
#define KPAD 4512
#define NTOT (NIMG * KPAD)
#define SORTN 8192
#define POSTN 1000
#define IMGW_F 800.0f
#define IMGH_F 800.0f
#define NMS_THR 0.7f
#define MIN_SZ 1e-3f
#define BBOX_CLIP_F 4.135166556742356f

__constant__ int c_NPL[NLEV]  = {120000, 30000, 7500, 1875, 507};   // A*H*W
__constant__ int c_OFF[NLEV]  = {0, 120000, 150000, 157500, 159375};
__constant__ int c_K[NLEV]    = {1000, 1000, 1000, 1000, 507};
__constant__ int c_KOFF[NLEV] = {0, 1000, 2000, 3000, 4000};
__constant__ int c_HW[NLEV]   = {40000, 10000, 2500, 625, 169};     // H*W

// ---------- CDNA5 async global->LDS (ASYNCcnt path) ----------
__device__ __forceinline__ void g2l_async_b128(void* lds_dst, unsigned long long gbase,
                                               unsigned voff_bytes) {
  // low 32 bits of a flat LDS pointer == LDS byte offset (aperture low bits are 0)
  unsigned lo = (unsigned)(uintptr_t)lds_dst;
  asm volatile("global_load_async_to_lds_b128 %0, %1, %2 offset:0"
               :: "v"(lo), "v"(voff_bytes), "s"(gbase)
               : "memory");
}
__device__ __forceinline__ void wait_async0() {
  asm volatile("s_wait_asynccnt 0x0" ::: "memory");
}
__device__ __forceinline__ void wait_async1() {     // allow 1 in flight (double buffer)
  asm volatile("s_wait_asynccnt 0x1" ::: "memory");
}

// monotonic float->uint key: larger float => larger uint
__device__ __forceinline__ unsigned f2key(float v) {
  unsigned u = __float_as_uint(v);
  return u ^ (unsigned)(((int)u >> 31) | 0x80000000);
}

// ================= Kernel A: per-(image,level) radix-select top-k =================
extern "C" __global__ void __launch_bounds__(1024)
rpn_topk(const float* __restrict__ o0, const float* __restrict__ o1,
         const float* __restrict__ o2, const float* __restrict__ o3,
         const float* __restrict__ o4,
         float* __restrict__ topVal, int* __restrict__ topIdx)
{
  __shared__ __align__(16) float stage[2][4096];   // double-buffered per-lane 16B slots
  __shared__ unsigned hist[256];
  __shared__ int sh_b, sh_k, sGt, sEq;

  const int tid = threadIdx.x;
  const int n = blockIdx.x / NLEV;
  const int l = blockIdx.x % NLEV;
  const float* objs[NLEV] = {o0, o1, o2, o3, o4};
  const int cnt = c_NPL[l];
  const int k   = c_K[l];
  const float* src = objs[l] + (long long)n * cnt;

  // 16B-align the quad region for b128 async loads
  unsigned long long addr = (unsigned long long)(uintptr_t)src;
  int head = (int)(((16u - (unsigned)(addr & 15ull)) & 15u) >> 2);
  if (head > cnt) head = cnt;
  const int nQuad = (cnt - head) >> 2;

  unsigned hiMask = 0u, hiVal = 0u;
  int kRem = k;

  for (int d = 3; d >= 0; --d) {
    for (int b = tid; b < 256; b += 1024) hist[b] = 0u;
    __syncthreads();
    // head + tail handled scalar
    for (int i = tid; i < head; i += 1024) {
      unsigned key = f2key(src[i]);
      if ((key & hiMask) == hiVal) atomicAdd(&hist[(key >> (8 * d)) & 255u], 1u);
    }
    for (int i = head + nQuad * 4 + tid; i < cnt; i += 1024) {
      unsigned key = f2key(src[i]);
      if ((key & hiMask) == hiVal) atomicAdd(&hist[(key >> (8 * d)) & 255u], 1u);
    }
    // bulk: double-buffered async global->LDS staging.
    // Depth-1 pipeline on the per-wave ASYNCcnt: prefetch next tile, wait <=1,
    // consume current; last iteration drains with wait 0.
    int buf = 0;
    if (tid < nQuad)
      g2l_async_b128(&stage[0][tid * 4], addr, (unsigned)((head + tid * 4) * 4));
    for (int q = tid; q < nQuad; q += 1024) {
      int qn = q + 1024;
      if (qn < nQuad) {
        g2l_async_b128(&stage[buf ^ 1][tid * 4], addr, (unsigned)((head + qn * 4) * 4));
        wait_async1();
      } else {
        wait_async0();
      }
      #pragma unroll
      for (int c = 0; c < 4; ++c) {
        unsigned key = f2key(stage[buf][tid * 4 + c]);
        if ((key & hiMask) == hiVal) atomicAdd(&hist[(key >> (8 * d)) & 255u], 1u);
      }
      buf ^= 1;
    }
    __syncthreads();
    if (tid == 0) {
      int acc = 0, bSel = 0, kNew = kRem;
      for (int v = 255; v >= 0; --v) {
        int h = (int)hist[v];
        if (acc + h >= kRem || v == 0) { bSel = v; kNew = kRem - acc; break; }
        acc += h;
      }
      sh_b = bSel; sh_k = kNew;
    }
    __syncthreads();
    hiVal |= ((unsigned)sh_b) << (8 * d);
    hiMask |= 255u << (8 * d);
    kRem = sh_k;
    __syncthreads();
  }

  if (tid == 0) { sGt = 0; sEq = 0; }
  __syncthreads();
  const unsigned T = hiVal;           // key of the k-th largest element
  const int outBase = n * KPAD + c_KOFF[l];
  const int hw = c_HW[l];
  for (int i = tid; i < cnt; i += 1024) {
    float v = src[i];
    unsigned key = f2key(v);
    int pos = -1;
    if (key > T) {
      pos = atomicAdd(&sGt, 1);
    } else if (key == T) {
      int q = atomicAdd(&sEq, 1);
      if (q < kRem) pos = (k - kRem) + q;
    }
    if (pos >= 0) {
      int a = i / hw;                 // source is (a, cell)-major
      int cell = i - a * hw;
      topVal[outBase + pos] = v;
      topIdx[outBase + pos] = c_OFF[l] + cell * 3 + a;   // flatten order (cell, a)
    }
  }
}

// ================= Kernel B: gather + decode + clip + validity =================
extern "C" __global__ void
rpn_decode(const float* __restrict__ d0, const float* __restrict__ d1,
           const float* __restrict__ d2, const float* __restrict__ d3,
           const float* __restrict__ d4, const float* __restrict__ anchors,
           const float* __restrict__ topVal, const int* __restrict__ topIdx,
           float* __restrict__ candBox, float* __restrict__ candKey,
           int* __restrict__ candLvl)
{
  int s = blockIdx.x * blockDim.x + threadIdx.x;
  if (s >= NTOT) return;
  int n = s / KPAD, p = s - n * KPAD;
  if (p >= KTOT) {          // padding slot: sinks below every real candidate
    candKey[s] = -2.0f; candLvl[s] = 0;
    candBox[4 * s + 0] = 0.f; candBox[4 * s + 1] = 0.f;
    candBox[4 * s + 2] = 0.f; candBox[4 * s + 3] = 0.f;
    return;
  }
  int l = (p < 1000) ? 0 : (p < 2000) ? 1 : (p < 3000) ? 2 : (p < 4000) ? 3 : 4;
  const float* dls[NLEV] = {d0, d1, d2, d3, d4};
  int g = topIdx[s];
  float logit = topVal[s];
  float score = 1.0f / (1.0f + __expf(-logit));
  int rel = g - c_OFF[l];
  int cell = rel / 3, a = rel - cell * 3;
  long long hw = (long long)c_HW[l];
  const float* dp = dls[l] + ((long long)n * 12 + a * 4) * hw + cell;
  __builtin_prefetch(dp, 0, 0);                 // global_prefetch_b8
  const float* an = anchors + (long long)g * 4;
  float dx = dp[0], dy = dp[hw], dw = dp[2 * hw], dh = dp[3 * hw];
  float ax1 = an[0], ay1 = an[1], ax2 = an[2], ay2 = an[3];
  float wa = ax2 - ax1, ha = ay2 - ay1;
  float cxa = ax1 + 0.5f * wa, cya = ay1 + 0.5f * ha;
  dw = fminf(dw, BBOX_CLIP_F);
  dh = fminf(dh, BBOX_CLIP_F);
  float cx = dx * wa + cxa, cy = dy * ha + cya;
  float w = __expf(dw) * wa, h = __expf(dh) * ha;
  float x1 = cx - 0.5f * w, y1 = cy - 0.5f * h;
  float x2 = cx + 0.5f * w, y2 = cy + 0.5f * h;
  x1 = fminf(fmaxf(x1, 0.f), IMGW_F);
  y1 = fminf(fmaxf(y1, 0.f), IMGH_F);
  x2 = fminf(fmaxf(x2, 0.f), IMGW_F);
  y2 = fminf(fmaxf(y2, 0.f), IMGH_F);
  bool valid = (x2 - x1 >= MIN_SZ) && (y2 - y1 >= MIN_SZ) && (score >= 0.0f);
  candBox[4 * s + 0] = x1; candBox[4 * s + 1] = y1;
  candBox[4 * s + 2] = x2; candBox[4 * s + 3] = y2;
  candKey[s] = valid ? score : -1.0f;   // ms = where(valid, score, -1)
  candLvl[s] = l;
}

// ================= Kernel C: per-image bitonic sort (desc) in LDS =================
extern "C" __global__ void __launch_bounds__(1024)
rpn_sort(const float* __restrict__ candKey, int* __restrict__ sortedSlot)
{
  __shared__ __align__(16) float sk[SORTN];
  __shared__ int sp[SORTN];
  const int n = blockIdx.x, tid = threadIdx.x;
  unsigned long long gb = (unsigned long long)(uintptr_t)(candKey + n * KPAD);
  // bulk async global->LDS copy of the 4512 sort keys (KPAD*4 bytes, 16B aligned)
  for (int j4 = tid * 4; j4 < KPAD; j4 += 4096)
    g2l_async_b128(&sk[j4], gb, (unsigned)(j4 * 4));
  for (int j = tid; j < SORTN; j += 1024) sp[j] = (j < KPAD) ? j : 0;
  for (int j = KPAD + tid; j < SORTN; j += 1024) sk[j] = -3.0e38f;
  wait_async0();
  __syncthreads();

  for (int size = 2; size <= SORTN; size <<= 1) {
    for (int stride = size >> 1; stride > 0; stride >>= 1) {
      __syncthreads();
      for (int t = tid; t < SORTN / 2; t += 1024) {
        int lo = ((t & ~(stride - 1)) << 1) | (t & (stride - 1));
        int hi = lo | stride;
        bool descBlk = (lo & size) == 0;
        float a = sk[lo], b = sk[hi];
        if ((a < b) == descBlk) {
          sk[lo] = b; sk[hi] = a;
          int tmp = sp[lo]; sp[lo] = sp[hi]; sp[hi] = tmp;
        }
      }
    }
  }
  __syncthreads();
  for (int j = tid; j < KPAD; j += 1024) sortedSlot[n * KPAD + j] = sp[j];
}

// ===== Kernel D: tiled greedy NMS (wave32 intra-chunk) + compaction + output =====
extern "C" __global__ void __launch_bounds__(1024)
rpn_nms(const float* __restrict__ candBox, const float* __restrict__ candKey,
        const int* __restrict__ candLvl, const int* __restrict__ sortedSlot,
        float* __restrict__ outBox, float* __restrict__ outScore)
{
  constexpr int SN = KPAD;
  __shared__ float sx1[SN], sy1[SN], sx2[SN], sy2[SN], sar[SN], skey[SN];
  __shared__ unsigned char slvl[SN], ssup[SN], skeep[SN];
  __shared__ int scanA[1024], scanB[1024];
  __shared__ unsigned chunkKeep;

  const int n = blockIdx.x, tid = threadIdx.x;

  // zero the output block (harness poisons d_out only once)
  for (int r = tid; r < POSTN; r += 1024) {
    outScore[n * POSTN + r] = 0.f;
    #pragma unroll
    for (int c = 0; c < 4; ++c) outBox[(n * POSTN + r) * 4 + c] = 0.f;
  }

  // gather candidates in sorted order into LDS (whole image fits: ~130KB < 320KB)
  for (int j = tid; j < SN; j += 1024) {
    int slot = sortedSlot[n * KPAD + j];
    int ab = n * KPAD + slot;
    float x1 = candBox[4 * ab + 0], y1 = candBox[4 * ab + 1];
    float x2 = candBox[4 * ab + 2], y2 = candBox[4 * ab + 3];
    sx1[j] = x1; sy1[j] = y1; sx2[j] = x2; sy2[j] = y2;
    sar[j] = (x2 - x1) * (y2 - y1);
    skey[j] = candKey[ab];
    slvl[j] = (unsigned char)candLvl[ab];
    ssup[j] = 0; skeep[j] = 0;
  }
  __syncthreads();

  // greedy NMS over 32-wide chunks:
  //   phase 1: wave 0 resolves the 32x32 triangle in lockstep (shuffles only)
  //   phase 2: all waves suppress the tail against the chunk's kept boxes
  // (level-offset trick == same-level-only suppression after clipping to image)
  for (int c0 = 0; c0 < KTOT; c0 += 32) {
    if (skey[c0] <= 0.0f) break;      // sorted desc: rest are invalid/padding
    const int cs = (KTOT - c0 < 32) ? (KTOT - c0) : 32;

    if (tid < 32) {                   // wave 0 only
      const int lane = tid;
      const int j = c0 + lane;
      const bool inb = lane < cs;
      int act = (inb && skey[j] > 0.0f && ssup[j] == 0) ? 1 : 0;
      float mx1 = 0, my1 = 0, mx2 = 0, my2 = 0, mar = 0; int mlv = -1;
      if (inb) { mx1 = sx1[j]; my1 = sy1[j]; mx2 = sx2[j]; my2 = sy2[j];
                 mar = sar[j]; mlv = slvl[j]; }
      unsigned keepBits = 0u;
      for (int i = 0; i < cs; ++i) {
        int actI = __shfl(act, i);    // ds_bpermute broadcast of lane i's state
        if (actI) {
          keepBits |= (1u << i);
          if (act && lane > i && (int)slvl[c0 + i] == mlv) {
            float ix = fminf(mx2, sx2[c0 + i]) - fmaxf(mx1, sx1[c0 + i]);
            float iy = fminf(my2, sy2[c0 + i]) - fmaxf(my1, sy1[c0 + i]);
            float inter = fmaxf(ix, 0.f) * fmaxf(iy, 0.f);
            float iou = inter / (mar + sar[c0 + i] - inter);
            if (iou > NMS_THR) act = 0;
          }
        }
      }
      if (inb) skeep[j] = (unsigned char)((keepBits >> lane) & 1u);
      if (lane == 0) chunkKeep = keepBits;
    }
    __syncthreads();

    const unsigned kb = chunkKeep;
    if (kb) {
      for (int j = c0 + 32 + tid; j < KTOT; j += 1024) {
        if (ssup[j] || skey[j] <= 0.0f) continue;
        float x1 = sx1[j], y1 = sy1[j], x2 = sx2[j], y2 = sy2[j], ar = sar[j];
        int lj = slvl[j];
        unsigned m = kb; bool sup = false;
        while (m) {
          int i = __ffs(m) - 1; m &= (m - 1);
          if ((int)slvl[c0 + i] != lj) continue;
          float ix = fminf(x2, sx2[c0 + i]) - fmaxf(x1, sx1[c0 + i]);
          float iy = fminf(y2, sy2[c0 + i]) - fmaxf(y1, sy1[c0 + i]);
          float inter = fmaxf(ix, 0.f) * fmaxf(iy, 0.f);
          float iou = inter / (ar + sar[c0 + i] - inter);
          if (iou > NMS_THR) { sup = true; break; }
        }
        if (sup) ssup[j] = 1;
      }
    }
    __syncthreads();
  }

  // block-wide exclusive prefix sum of keep flags (chunked + Hillis-Steele)
  const int CH = 5;                    // 1024*5 >= KTOT
  int start = tid * CH, cnt = 0;
  for (int e = 0; e < CH; ++e) {
    int j = start + e;
    if (j < KTOT) cnt += skeep[j];
  }
  scanA[tid] = cnt;
  __syncthreads();
  int* cur = scanA; int* nxt = scanB;
  for (int off = 1; off < 1024; off <<= 1) {
    int v = cur[tid];
    if (tid >= off) v += cur[tid - off];
    nxt[tid] = v;
    __syncthreads();
    int* t_ = cur; cur = nxt; nxt = t_;
  }
  int rank = (tid == 0) ? 0 : cur[tid - 1];
  for (int e = 0; e < CH; ++e) {
    int j = start + e;
    if (j >= KTOT) break;
    if (skeep[j]) {
      if (rank < POSTN) {
        outScore[n * POSTN + rank] = skey[j];
        outBox[(n * POSTN + rank) * 4 + 0] = sx1[j];
        outBox[(n * POSTN + rank) * 4 + 1] = sy1[j];
        outBox[(n * POSTN + rank) * 4 + 2] = sx2[j];
        outBox[(n * POSTN + rank) * 4 + 3] = sy2[j];
      }
      ++rank;
    }
  }
}

// ================================ launch ================================
extern "C" void kernel_launch(void* const* d_in, const int* in_sizes, int n_in,
                              void* d_out, int out_size, void* d_ws, size_t ws_size,
                              hipStream_t stream)
{
  (void)in_sizes; (void)n_in; (void)out_size; (void)ws_size;
  const float* obj[NLEV]; const float* del[NLEV];
  for (int i = 0; i < NLEV; ++i) {
    obj[i] = (const float*)d_in[2 * i];
    del[i] = (const float*)d_in[2 * i + 1];
  }
  const float* anchors = (const float*)d_in[10];

  char* w = (char*)d_ws;
  float* topVal     = (float*)w; w += (size_t)NTOT * sizeof(float);
  int*   topIdx     = (int*)w;   w += (size_t)NTOT * sizeof(int);
  float* candBox    = (float*)w; w += (size_t)NTOT * 4 * sizeof(float);
  float* candKey    = (float*)w; w += (size_t)NTOT * sizeof(float);
  int*   candLvl    = (int*)w;   w += (size_t)NTOT * sizeof(int);
  int*   sortedSlot = (int*)w;   w += (size_t)NTOT * sizeof(int);

  float* outBox   = (float*)d_out;
  float* outScore = outBox + (size_t)NIMG * POSTN * 4;

  rpn_topk<<<NIMG * NLEV, 1024, 0, stream>>>(obj[0], obj[1], obj[2], obj[3], obj[4],
                                             topVal, topIdx);
  rpn_decode<<<(NTOT + 255) / 256, 256, 0, stream>>>(del[0], del[1], del[2], del[3], del[4],
                                                     anchors, topVal, topIdx,
                                                     candBox, candKey, candLvl);
  rpn_sort<<<NIMG, 1024, 0, stream>>>(candKey, sortedSlot);
  rpn_nms<<<NIMG, 1024, 0, stream>>>(candBox, candKey, candLvl, sortedSlot,
                                     outBox, outScore);
}